// AdaptiveGCNLayer_73624329388096
// MI455X (gfx1250) — compile-verified
//
#include <hip/hip_runtime.h>

typedef __attribute__((ext_vector_type(2))) float v2f;
typedef __attribute__((ext_vector_type(8))) float v8f;

static constexpr int NB   = 8192;   // frames
static constexpr int NV   = 25;     // nodes per frame
static constexpr int NVP  = 32;     // padded rows (M padded to 2x16 for WMMA)
static constexpr int NC   = 128;    // channels (C_IN == C_OUT)
static constexpr int NE   = 48;     // edges per frame
static constexpr int FPB  = 8;      // frames per block
static constexpr int XSTR = 132;    // LDS row stride (words), rows 16B aligned
static constexpr int WSTR = 130;    // LDS row stride (words) for transposed W, rows 8B aligned

// ws float-word layout: [0..47] norm, [64..88] invdeg, int words [96..143] src, [160..207] tgt
static constexpr int WS_NORM = 0;
static constexpr int WS_IDEG = 64;
static constexpr int WS_SRC  = 96;
static constexpr int WS_TGT  = 160;

// ---------------------------------------------------------------------------
// Prep: edge structure is identical for every frame -> compute norms once.
// ---------------------------------------------------------------------------
__global__ __launch_bounds__(64) void gcn_prep(const long long* __restrict__ ei,
                                               float* __restrict__ wsf,
                                               int* __restrict__ wsi) {
  const int t = threadIdx.x;
  if (t < NE) {
    const int s = (int)ei[t];
    const int d = (int)ei[NE + t];
    wsi[WS_SRC + t] = s;
    wsi[WS_TGT + t] = d;
    int degS = 1, degD = 1;                 // self-loop contributes 1
    for (int e = 0; e < NE; ++e) {
      const int te = (int)ei[NE + e];
      degS += (te == s);
      degD += (te == d);
    }
    wsf[WS_NORM + t] = rsqrtf((float)degS) * rsqrtf((float)degD);
  }
  if (t < NV) {
    int deg = 1;
    for (int e = 0; e < NE; ++e) deg += ((int)ei[NE + e] == t);
    wsf[WS_IDEG + t] = 1.0f / (float)deg;   // self-loop term: dis[n]^2 = 1/deg
  }
}

// ---------------------------------------------------------------------------
// Fused per-frame: xw = x_f @ W via V_WMMA_F32_16X16X4_F32, then graph
// aggregation entirely in LDS (ds_add_f32), biased store to global.
// ---------------------------------------------------------------------------
__global__ __launch_bounds__(256) void gcn_main(const float* __restrict__ x,
                                                const float* __restrict__ W,
                                                const float* __restrict__ bias,
                                                const float* __restrict__ wsf,
                                                const int* __restrict__ wsi,
                                                float* __restrict__ out) {
  __shared__ __attribute__((aligned(16))) float Wt[NC * WSTR];    // W transposed: Wt[n][k]
  __shared__ __attribute__((aligned(16))) float xs[NVP * XSTR];   // staged x frame (rows 25..31 = dead pad)
  __shared__ __attribute__((aligned(16))) float xw[NVP * XSTR];   // GEMM result (rows 25..31 = dead pad)
  __shared__ __attribute__((aligned(16))) float ob[NV * XSTR];    // aggregation buffer
  __shared__ float nrm[NE];
  __shared__ float idg[NV];
  __shared__ int   esrc[NE];
  __shared__ int   etgt[NE];

  const int tid  = threadIdx.x;
  const int lane = tid & 31;
  const int wv   = tid >> 5;      // wave id 0..7 -> 16-column strip of output
  const int hh   = lane >> 4;     // half-wave select
  const int lm   = lane & 15;

  // stage small tables
  if (tid < NE) { nrm[tid] = wsf[WS_NORM + tid]; esrc[tid] = wsi[WS_SRC + tid]; etgt[tid] = wsi[WS_TGT + tid]; }
  if (tid < NV) { idg[tid] = wsf[WS_IDEG + tid]; }

  // stage W transposed into LDS (once, reused for FPB frames)
  for (int j = tid; j < (NC * NC) / 4; j += 256) {
    const int k  = j >> 5;
    const int n0 = (j & 31) << 2;
    const float4 w4 = *(const float4*)(W + k * NC + n0);
    Wt[(n0 + 0) * WSTR + k] = w4.x;
    Wt[(n0 + 1) * WSTR + k] = w4.y;
    Wt[(n0 + 2) * WSTR + k] = w4.z;
    Wt[(n0 + 3) * WSTR + k] = w4.w;
  }

  const int f0 = blockIdx.x * FPB;
  for (int f = f0; f < f0 + FPB; ++f) {
    const float* xg = x + (size_t)f * (NV * NC);

    // stage x frame (coalesced b128 loads)
    for (int j = tid; j < (NV * NC) / 4; j += 256) {
      const int v  = j >> 5;
      const int c0 = (j & 31) << 2;
      *(float4*)(xs + v * XSTR + c0) = *(const float4*)(xg + v * NC + c0);
    }
    // prefetch next frame's x (global_prefetch_b8) to overlap with compute
    if (f + 1 < f0 + FPB && tid < 100)
      __builtin_prefetch((const void*)(xg + NV * NC + tid * 32), 0, 0);
    __syncthreads();

    // ---- GEMM: xw[32x128] = xs[32x128] @ W[128x128] (rows 25..31 dead) ----
    // WMMA row independence: garbage in pad rows of A only pollutes pad rows
    // of D, which are never read -> no guards, EXEC stays all-ones.
    v8f acc0 = {};            // rows 0..15
    v8f acc1 = {};            // rows 16..31
    const int    ncol = (wv << 4) + lm;            // this lane's output column
    const float* btab  = &Wt[ncol * XSTR - ncol * (XSTR - WSTR)]; // = &Wt[ncol*WSTR]
    const float* arow0 = xs + lm * XSTR;
    const float* arow1 = xs + (16 + lm) * XSTR;
#pragma unroll 4
    for (int kk = 0; kk < NC; kk += 4) {
      const int kb = kk + hh * 2;                  // per-lane K pair
      const v2f a0 = *(const v2f*)(arow0 + kb);
      const v2f a1 = *(const v2f*)(arow1 + kb);
      const v2f bf = *(const v2f*)(btab + kb);
      acc0 = __builtin_amdgcn_wmma_f32_16x16x4_f32(false, a0, false, bf, (short)0,
                                                   acc0, false, false);
      acc1 = __builtin_amdgcn_wmma_f32_16x16x4_f32(false, a1, false, bf, (short)0,
                                                   acc1, false, false);
    }
    // spill accumulators to LDS (C/D layout: VGPR r -> row r + 8*half),
    // unconditional: rows 25..31 land in xw's dead pad.
#pragma unroll
    for (int r = 0; r < 8; ++r) {
      xw[(r + hh * 8) * XSTR + ncol]      = acc0[r];
      xw[(16 + r + hh * 8) * XSTR + ncol] = acc1[r];
    }
    __syncthreads();

    // ---- aggregation: self-loop init ----
    for (int j = tid; j < NV * NC; j += 256) {
      const int v = j >> 7, c = j & 127;
      ob[v * XSTR + c] = xw[v * XSTR + c] * idg[v];
    }
    __syncthreads();

    // ---- edge scatter via LDS float atomics (ds_add_f32) ----
    for (int j = tid; j < NE * NC; j += 256) {
      const int e = j >> 7, c = j & 127;
      atomicAdd(&ob[etgt[e] * XSTR + c], xw[esrc[e] * XSTR + c] * nrm[e]);
    }
    __syncthreads();

    // ---- biased writeback (coalesced) ----
    float* og = out + (size_t)f * (NV * NC);
    for (int j = tid; j < NV * NC; j += 256) {
      const int c = j & 127;
      og[j] = ob[(j >> 7) * XSTR + c] + bias[c];
    }
    __syncthreads();
  }
}

// ---------------------------------------------------------------------------
extern "C" void kernel_launch(void* const* d_in, const int* in_sizes, int n_in,
                              void* d_out, int out_size, void* d_ws, size_t ws_size,
                              hipStream_t stream) {
  const float*     xp = (const float*)d_in[0];     // x [8192,25,128]
  const long long* ei = (const long long*)d_in[1]; // edge_index [2,48] int64
  // d_in[2] adj_matrix, d_in[3] aw_W, d_in[4] aw_b feed only the dead
  // "adaptive adjacency" branch of the reference -> intentionally unused.
  const float* gw = (const float*)d_in[5];         // gcn_W [128,128]
  const float* gb = (const float*)d_in[6];         // gcn_b [128]
  float* outp = (float*)d_out;
  float* wsf  = (float*)d_ws;
  int*   wsi  = (int*)d_ws;

  gcn_prep<<<1, 64, 0, stream>>>(ei, wsf, wsi);
  gcn_main<<<NB / FPB, 256, 0, stream>>>(xp, gw, gb, wsf, wsi, outp);
}